// _EMOptFunc_33818572488944
// MI455X (gfx1250) — compile-verified
//
#include <hip/hip_runtime.h>
#include <math.h>

// CDNA5 / gfx1250. wave32. WMMA f32 16x16x4 used for the alpha-weighted
// covariance reductions (exact f32 math; layout-invariant construction:
// A = all-ones so only the documented "col = lane&15" striping matters).

typedef float v2f __attribute__((ext_vector_type(2)));
typedef float v8f __attribute__((ext_vector_type(8)));

#define TPB 256
#define NB  512

// workspace layout (float offsets)
#define WS_PINV 0                 // 16 floats: prior_inv (full symmetric 4x4)
#define WS_GMAX 16                // 1 float : global max of s
#define WS_BMAX 32                // NB floats: per-block max
#define WS_PZ   (32 + NB)         // NB floats: per-block sum of weights
#define WS_PW   (32 + 2 * NB)     // NB*16 floats: per-block weighted sums

struct Chol {
  float L10, L20, L30, L21, L31, L32;
  float i0, i1, i2, i3;      // 1/L00, 1/L11, 1/L22, 1/L33
  float logdetP;
};

__device__ __forceinline__ Chol chol4(float P00, float P01, float P02, float P03,
                                      float P11, float P12, float P13,
                                      float P22, float P23, float P33) {
  Chol c;
  float L00 = sqrtf(P00);                      c.i0 = 1.0f / L00;
  c.L10 = P01 * c.i0; c.L20 = P02 * c.i0; c.L30 = P03 * c.i0;
  float L11 = sqrtf(P11 - c.L10 * c.L10);      c.i1 = 1.0f / L11;
  c.L21 = (P12 - c.L20 * c.L10) * c.i1;
  c.L31 = (P13 - c.L30 * c.L10) * c.i1;
  float L22 = sqrtf(P22 - c.L20 * c.L20 - c.L21 * c.L21); c.i2 = 1.0f / L22;
  c.L32 = (P23 - c.L30 * c.L20 - c.L31 * c.L21) * c.i2;
  float L33 = sqrtf(P33 - c.L30 * c.L30 - c.L31 * c.L31 - c.L32 * c.L32);
  c.i3 = 1.0f / L33;
  c.logdetP = 2.0f * logf(L00 * L11 * L22 * L33);
  return c;
}

// K0: one thread inverts the 4x4 prior (SPD) -> ws[WS_PINV..+15]
__global__ void k0_priorinv(const float* ec, const int* liter, float* ws) {
  if (threadIdx.x != 0 || blockIdx.x != 0) return;
  const float* Pp = ec + (*liter) * 16;
  Chol c = chol4(Pp[0], Pp[1], Pp[2], Pp[3], Pp[5], Pp[6], Pp[7], Pp[10], Pp[11], Pp[15]);
  float Li00 = c.i0, Li11 = c.i1, Li22 = c.i2, Li33 = c.i3;
  float Li10 = -c.i1 * (c.L10 * Li00);
  float Li21 = -c.i2 * (c.L21 * Li11);
  float Li20 = -c.i2 * (c.L20 * Li00 + c.L21 * Li10);
  float Li32 = -c.i3 * (c.L32 * Li22);
  float Li31 = -c.i3 * (c.L31 * Li11 + c.L32 * Li21);
  float Li30 = -c.i3 * (c.L30 * Li00 + c.L31 * Li10 + c.L32 * Li20);
  float C00 = Li00*Li00 + Li10*Li10 + Li20*Li20 + Li30*Li30;
  float C01 = Li10*Li11 + Li20*Li21 + Li30*Li31;
  float C02 = Li20*Li22 + Li30*Li32;
  float C03 = Li30*Li33;
  float C11 = Li11*Li11 + Li21*Li21 + Li31*Li31;
  float C12 = Li21*Li22 + Li31*Li32;
  float C13 = Li31*Li33;
  float C22 = Li22*Li22 + Li32*Li32;
  float C23 = Li32*Li33;
  float C33 = Li33*Li33;
  float o[16] = {C00,C01,C02,C03, C01,C11,C12,C13, C02,C12,C22,C23, C03,C13,C23,C33};
  for (int i = 0; i < 16; ++i) ws[WS_PINV + i] = o[i];
}

// K1: s_p = log(pi_p) + 0.5*(||L^-1 r||^2 - logdet(precision)); per-block max.
__global__ void k1_smax(const float4* __restrict__ r4, const float4* __restrict__ q4,
                        const float* __restrict__ pi, float* ws, int P) {
  float pv[16];
#pragma unroll
  for (int i = 0; i < 16; ++i) pv[i] = ws[WS_PINV + i];
  float lmax = -INFINITY;
  for (int p = blockIdx.x * TPB + threadIdx.x; p < P; p += TPB * NB) {
    float4 q0 = q4[p*4+0], q1 = q4[p*4+1], q2 = q4[p*4+2], q3 = q4[p*4+3];
    Chol c = chol4(q0.x+pv[0], q0.y+pv[1], q0.z+pv[2], q0.w+pv[3],
                   q1.y+pv[5], q1.z+pv[6], q1.w+pv[7],
                   q2.z+pv[10], q2.w+pv[11], q3.w+pv[15]);
    float4 rv = r4[p];
    float y0 = rv.x * c.i0;
    float y1 = (rv.y - c.L10*y0) * c.i1;
    float y2 = (rv.z - c.L20*y0 - c.L21*y1) * c.i2;
    float y3 = (rv.w - c.L30*y0 - c.L31*y1 - c.L32*y2) * c.i3;
    float quad = y0*y0 + y1*y1 + y2*y2 + y3*y3;
    float s = logf(pi[p]) + 0.5f * (quad - c.logdetP);
    lmax = fmaxf(lmax, s);
  }
  for (int off = 16; off > 0; off >>= 1) lmax = fmaxf(lmax, __shfl_down(lmax, off, 32));
  __shared__ float sm[TPB / 32];
  if ((threadIdx.x & 31) == 0) sm[threadIdx.x >> 5] = lmax;
  __syncthreads();
  if (threadIdx.x == 0) {
    float m = sm[0];
    for (int i = 1; i < TPB / 32; ++i) m = fmaxf(m, sm[i]);
    ws[WS_BMAX + blockIdx.x] = m;
  }
}

// K1b: reduce NB block maxes -> global max
__global__ void k1b_gmax(float* ws) {
  __shared__ float sm[NB];
  int t = threadIdx.x;
  sm[t] = ws[WS_BMAX + t];
  __syncthreads();
  for (int off = NB / 2; off > 0; off >>= 1) {
    if (t < off) sm[t] = fmaxf(sm[t], sm[t + off]);
    __syncthreads();
  }
  if (t == 0) ws[WS_GMAX] = sm[0];
}

// K2: w = exp(s-M); acc[16] += w*(post_covar + m m^T); block-reduce the 16-wide
// accumulators with chained V_WMMA_F32_16X16X4_F32 (A = ones sums 4 staged rows).
__global__ void k2_accum(const float4* __restrict__ r4, const float4* __restrict__ q4,
                         const float* __restrict__ pi, float* ws, int P) {
  float pv[16];
#pragma unroll
  for (int i = 0; i < 16; ++i) pv[i] = ws[WS_PINV + i];
  const float M = ws[WS_GMAX];

  float acc[16];
#pragma unroll
  for (int i = 0; i < 16; ++i) acc[i] = 0.0f;
  float accZ = 0.0f;

  for (int p = blockIdx.x * TPB + threadIdx.x; p < P; p += TPB * NB) {
    float4 q0 = q4[p*4+0], q1 = q4[p*4+1], q2 = q4[p*4+2], q3 = q4[p*4+3];
    Chol c = chol4(q0.x+pv[0], q0.y+pv[1], q0.z+pv[2], q0.w+pv[3],
                   q1.y+pv[5], q1.z+pv[6], q1.w+pv[7],
                   q2.z+pv[10], q2.w+pv[11], q3.w+pv[15]);
    float4 rv = r4[p];
    float y0 = rv.x * c.i0;
    float y1 = (rv.y - c.L10*y0) * c.i1;
    float y2 = (rv.z - c.L20*y0 - c.L21*y1) * c.i2;
    float y3 = (rv.w - c.L30*y0 - c.L31*y1 - c.L32*y2) * c.i3;
    float quad = y0*y0 + y1*y1 + y2*y2 + y3*y3;
    float s = logf(pi[p]) + 0.5f * (quad - c.logdetP);
    float w = expf(s - M);
    // back-substitution: m = precision^-1 r
    float m3 = y3 * c.i3;
    float m2 = (y2 - c.L32*m3) * c.i2;
    float m1 = (y1 - c.L21*m2 - c.L31*m3) * c.i1;
    float m0 = (y0 - c.L10*m1 - c.L20*m2 - c.L30*m3) * c.i0;
    // post_covar = Li^T Li
    float Li00 = c.i0, Li11 = c.i1, Li22 = c.i2, Li33 = c.i3;
    float Li10 = -c.i1 * (c.L10 * Li00);
    float Li21 = -c.i2 * (c.L21 * Li11);
    float Li20 = -c.i2 * (c.L20 * Li00 + c.L21 * Li10);
    float Li32 = -c.i3 * (c.L32 * Li22);
    float Li31 = -c.i3 * (c.L31 * Li11 + c.L32 * Li21);
    float Li30 = -c.i3 * (c.L30 * Li00 + c.L31 * Li10 + c.L32 * Li20);
    float Cm[4][4];
    Cm[0][0] = Li00*Li00 + Li10*Li10 + Li20*Li20 + Li30*Li30;
    Cm[0][1] = Li10*Li11 + Li20*Li21 + Li30*Li31;
    Cm[0][2] = Li20*Li22 + Li30*Li32;
    Cm[0][3] = Li30*Li33;
    Cm[1][1] = Li11*Li11 + Li21*Li21 + Li31*Li31;
    Cm[1][2] = Li21*Li22 + Li31*Li32;
    Cm[1][3] = Li31*Li33;
    Cm[2][2] = Li22*Li22 + Li32*Li32;
    Cm[2][3] = Li32*Li33;
    Cm[3][3] = Li33*Li33;
    Cm[1][0] = Cm[0][1]; Cm[2][0] = Cm[0][2]; Cm[3][0] = Cm[0][3];
    Cm[2][1] = Cm[1][2]; Cm[3][1] = Cm[1][3]; Cm[3][2] = Cm[2][3];
    float mv[4] = {m0, m1, m2, m3};
#pragma unroll
    for (int i = 0; i < 4; ++i)
#pragma unroll
      for (int j = 0; j < 4; ++j)
        acc[i*4+j] = fmaf(w, Cm[i][j] + mv[i]*mv[j], acc[i*4+j]);
    accZ += w;
  }

  __shared__ float sacc[TPB * 16];
  __shared__ float szz[TPB / 32];
#pragma unroll
  for (int j = 0; j < 16; ++j) sacc[threadIdx.x * 16 + j] = acc[j];
  for (int off = 16; off > 0; off >>= 1) accZ += __shfl_down(accZ, off, 32);
  if ((threadIdx.x & 31) == 0) szz[threadIdx.x >> 5] = accZ;
  __syncthreads();

  if (threadIdx.x < 32) {              // wave 0, EXEC all ones (uniform branch)
    v2f a = {1.0f, 1.0f};
    v8f cacc = {0.f, 0.f, 0.f, 0.f, 0.f, 0.f, 0.f, 0.f};
    int half = threadIdx.x >> 4, col = threadIdx.x & 15;
    for (int rowc = 0; rowc < TPB; rowc += 4) {
      v2f b;
      b[0] = sacc[(rowc + 2*half + 0) * 16 + col];
      b[1] = sacc[(rowc + 2*half + 1) * 16 + col];
      cacc = __builtin_amdgcn_wmma_f32_16x16x4_f32(false, a, false, b,
                                                   (short)0, cacc, false, false);
    }
    float z = 0.0f;
    for (int i = 0; i < TPB / 32; ++i) z += szz[i];
    if (threadIdx.x < 16) ws[WS_PW + blockIdx.x * 16 + threadIdx.x] = cacc[0];
    if (threadIdx.x == 0) ws[WS_PZ + blockIdx.x] = z;
  }
}

// K3: fold NB partials with chained WMMA, normalize, emit full effect_covar.
__global__ void k3_final(const float* ec, const int* liter, const float* ws,
                         float* out, int outN) {
  int t = threadIdx.x;
  if (t < outN) out[t] = ec[t];
  __syncthreads();
  if (t < 32) {
    const float* pw = ws + WS_PW;
    const float* pz = ws + WS_PZ;
    v2f a = {1.0f, 1.0f};
    v8f cacc = {0.f, 0.f, 0.f, 0.f, 0.f, 0.f, 0.f, 0.f};
    int half = t >> 4, col = t & 15;
    for (int rowc = 0; rowc < NB; rowc += 4) {
      v2f b;
      b[0] = pw[(rowc + 2*half + 0) * 16 + col];
      b[1] = pw[(rowc + 2*half + 1) * 16 + col];
      cacc = __builtin_amdgcn_wmma_f32_16x16x4_f32(false, a, false, b,
                                                   (short)0, cacc, false, false);
    }
    float z = 0.0f;
    for (int i = t; i < NB; i += 32) z += pz[i];
    for (int off = 16; off > 0; off >>= 1) z += __shfl_down(z, off, 32);
    z = __shfl(z, 0, 32);
    if (t < 16) out[(*liter) * 16 + t] = cacc[0] / z;
  }
}

extern "C" void kernel_launch(void* const* d_in, const int* in_sizes, int n_in,
                              void* d_out, int out_size, void* d_ws, size_t ws_size,
                              hipStream_t stream) {
  const float* rTZ = (const float*)d_in[0];   // (P,4)
  const float* is2 = (const float*)d_in[1];   // (P,4,4)
  const float* pi  = (const float*)d_in[2];   // (P,)
  const float* ec  = (const float*)d_in[3];   // (L,4,4)
  const int* liter = (const int*)d_in[4];     // scalar
  float* out = (float*)d_out;
  float* ws  = (float*)d_ws;
  int P = in_sizes[2];

  k0_priorinv<<<1, 1, 0, stream>>>(ec, liter, ws);
  k1_smax<<<NB, TPB, 0, stream>>>((const float4*)rTZ, (const float4*)is2, pi, ws, P);
  k1b_gmax<<<1, NB, 0, stream>>>(ws);
  k2_accum<<<NB, TPB, 0, stream>>>((const float4*)rTZ, (const float4*)is2, pi, ws, P);
  k3_final<<<1, 256, 0, stream>>>(ec, liter, ws, out, out_size);
}